// Weight_Schema_7928509628753
// MI455X (gfx1250) — compile-verified
//
#include <hip/hip_runtime.h>

// Problem constants (match the reference).
#define N_NODES 100000
#define HID     128
#define TILE_M  16
#define MTILES  (N_NODES / TILE_M)     // 6250 exactly — no remainder tiles
#define WAVES_PER_BLOCK 8

typedef __attribute__((ext_vector_type(2))) float v2f;   // A/B frag for wmma f32 16x16x4
typedef __attribute__((ext_vector_type(8))) float v8f;   // C/D frag (16x16 f32 / 32 lanes)

// ---------------------------------------------------------------------------
// Phase 1: hw = h @ W   via V_WMMA_F32_16X16X4_F32.
// One wave -> one 16-row tile x full 128 cols (8 n-tiles).
// weight (64 KB) staged in LDS once per 256-thread block, reused by 8 waves.
// ---------------------------------------------------------------------------
__global__ __launch_bounds__(256) void gemm_hw_kernel(
    const float* __restrict__ h,
    const float* __restrict__ w,
    float* __restrict__ hw) {
  __shared__ float wlds[HID * HID];           // weight, row-major [k][n], 64 KB

  const int t    = threadIdx.x;
  const int wave = t >> 5;
  const int lane = t & 31;

  // Cooperative coalesced stage of weight into LDS: 16384 floats / 256 thr.
#pragma unroll 4
  for (int i = 0; i < (HID * HID) / 256; ++i)
    wlds[i * 256 + t] = w[i * 256 + t];
  __syncthreads();

  const int mtile = blockIdx.x * WAVES_PER_BLOCK + wave;
  if (mtile >= MTILES) return;                // wave-uniform guard: EXEC all-1s inside

  const int lo = lane & 15;                   // M (for A) / N (for B,D)
  const int hi = lane >> 4;                   // K half-select per ISA layout

  v8f acc[8] = {};                            // 8 n-tiles of 16x16 f32 accumulators

  const float* arow = h + (size_t)(mtile * TILE_M + lo) * HID;

  for (int k = 0; k < HID; k += 4) {
    // A frag: lanes 0-15 -> K = k,k+1 ; lanes 16-31 -> K = k+2,k+3 (ISA 16x4 layout)
    v2f a;
    a.x = arow[k + 2 * hi];
    a.y = arow[k + 2 * hi + 1];
#pragma unroll
    for (int nt = 0; nt < 8; ++nt) {
      // B frag: B[K][N] with the same K half-split, N = nt*16 + lo
      v2f b;
      b.x = wlds[(k + 2 * hi) * HID + nt * 16 + lo];
      b.y = wlds[(k + 2 * hi + 1) * HID + nt * 16 + lo];
      // 8-arg form: (neg_a, A, neg_b, B, c_mod, C, reuse_a, reuse_b)
      acc[nt] = __builtin_amdgcn_wmma_f32_16x16x4_f32(
          false, a, false, b, (short)0, acc[nt], false, false);
    }
  }

  // D layout: VGPR v -> row (v + 8*hi), col = nt*16 + lo
  float* dst = hw + (size_t)mtile * TILE_M * HID;
#pragma unroll
  for (int nt = 0; nt < 8; ++nt) {
#pragma unroll
    for (int v = 0; v < 8; ++v)
      dst[(v + 8 * hi) * HID + nt * 16 + lo] = acc[nt][v];
  }
}

// ---------------------------------------------------------------------------
// Phase 2: COO SpMM. One wave per edge: 32 lanes x float4 = 128 channels.
// Coalesced 512B gather of hw[col], native f32 atomic scatter-add to out[row].
// hw + out (102 MB) are L2-resident (192 MB) -> L2/atomic-unit bound.
// ---------------------------------------------------------------------------
__global__ __launch_bounds__(256) void spmm_kernel(
    const float* __restrict__ hw,
    const float* __restrict__ vals,
    const int* __restrict__ rows,
    const int* __restrict__ cols,
    float* __restrict__ out,
    int nEdges) {
  const int edge = (int)((blockIdx.x * (unsigned)blockDim.x + threadIdx.x) >> 5);
  const int lane = threadIdx.x & 31;
  if (edge >= nEdges) return;

  // Wave-uniform scalars (force SGPR broadcast).
  const int   c = __builtin_amdgcn_readfirstlane(cols[edge]);
  const int   r = __builtin_amdgcn_readfirstlane(rows[edge]);
  const float v = vals[edge];

  const float4* src = reinterpret_cast<const float4*>(hw + (size_t)c * HID);
  float4 m = src[lane];

  float* dst = out + (size_t)r * HID + lane * 4;
  // unsafeAtomicAdd -> global_atomic_add_f32 (no CAS loop; RNE per CDNA5 L2 spec)
  unsafeAtomicAdd(dst + 0, m.x * v);
  unsafeAtomicAdd(dst + 1, m.y * v);
  unsafeAtomicAdd(dst + 2, m.z * v);
  unsafeAtomicAdd(dst + 3, m.w * v);
}

// ---------------------------------------------------------------------------
extern "C" void kernel_launch(void* const* d_in, const int* in_sizes, int n_in,
                              void* d_out, int out_size, void* d_ws, size_t ws_size,
                              hipStream_t stream) {
  const float* h     = (const float*)d_in[0];   // [N, HID]
  const float* w     = (const float*)d_in[1];   // [HID, HID]
  // d_in[2] = bias: reference discards tanh(out+bias); intentionally unused.
  const float* evals = (const float*)d_in[3];   // [K*E] flat
  const int*   erows = (const int*)  d_in[4];   // [K*E] flat
  const int*   ecols = (const int*)  d_in[5];   // [K*E] flat

  float* out   = (float*)d_out;
  float* hwbuf = (float*)d_ws;                  // N*HID*4 = 51.2 MB intermediate

  const int nEdges = in_sizes[3];               // K*E = 1,200,000

  // segment_sum starts from zero; d_out is poisoned by the harness.
  hipMemsetAsync(d_out, 0, (size_t)out_size * sizeof(float), stream);

  // Phase 1: GEMM. 6250 row-tiles / 8 waves per block.
  dim3 gemmGrid((MTILES + WAVES_PER_BLOCK - 1) / WAVES_PER_BLOCK);
  gemm_hw_kernel<<<gemmGrid, 256, 0, stream>>>(h, w, hwbuf);

  // Phase 2: SpMM. One wave per edge, 8 edges per 256-thread block.
  if (nEdges > 0) {
    dim3 spmmGrid((nEdges + 7) / 8);
    spmm_kernel<<<spmmGrid, 256, 0, stream>>>(hwbuf, evals, erows, ecols, out, nEdges);
  }
}